// QValueAgent_33681133535601
// MI455X (gfx1250) — compile-verified
//
#include <hip/hip_runtime.h>
#include <hip/hip_bf16.h>

#ifndef __has_builtin
#define __has_builtin(x) 0
#endif

#if __has_builtin(__builtin_amdgcn_global_load_async_to_lds_b128)
#define HAVE_ASYNC_B128 1
#else
#define HAVE_ASYNC_B128 0
#endif

#if __has_builtin(__builtin_amdgcn_s_wait_asynccnt)
#define HAVE_WAIT_ASYNC 1
#else
#define HAVE_WAIT_ASYNC 0
#endif

#define B_SESS   4096
#define T_TRIALS 2048
#define CH       64                  // timesteps per chunk (= per lane)
#define NC       32                  // chunks per session == wave32 lanes
#define NTASK    (B_SESS * NC)       // 131072 lane tasks

#define BLOCK          64                        // 2 waves / block
#define LANE_DATA_B    (CH * 12)                 // 768 B input per lane
#define LANE_STRIDE_B  784                       // 768 + 16 pad (16-aligned, 196 dw, gcd(196,64)=4)
#define LDS_BYTES      (BLOCK * LANE_STRIDE_B)   // 50176 B < 64 KB static limit
#define N_ASYNC        (LANE_DATA_B / 16)        // 48 b128 copies per lane

typedef int   v4i __attribute__((ext_vector_type(4)));
typedef float v4f __attribute__((ext_vector_type(4)));
typedef __attribute__((address_space(1))) v4i* gv4p;   // global (AS1)
typedef __attribute__((address_space(3))) v4i* lv4p;   // LDS    (AS3)

__device__ __forceinline__ void async_copy16(const void* gsrc, void* ldst) {
#if HAVE_ASYNC_B128
  __builtin_amdgcn_global_load_async_to_lds_b128((gv4p)gsrc, (lv4p)ldst, 0, 0);
#else
  typedef const __attribute__((address_space(1))) void* gptr_t;
  typedef __attribute__((address_space(3)))       void* lptr_t;
  gptr_t g = (gptr_t)gsrc;
  lptr_t l = (lptr_t)ldst;
  asm volatile("global_load_async_to_lds_b128 %0, %1, off"
               :: "v"(l), "v"(g) : "memory");
#endif
}

template <int N>
__device__ __forceinline__ void wait_async() {
#if HAVE_WAIT_ASYNC
  __builtin_amdgcn_s_wait_asynccnt(N);
#else
  asm volatile("s_wait_asynccnt %0" :: "n"(N) : "memory");
#endif
  asm volatile("" ::: "memory");   // fence: keep LDS reads below the wait
}

// sigmoid(alphas_raw) and alphas*k (uniform -> SALU float path on gfx1250)
__device__ __forceinline__ void load_params(const float* __restrict__ ar,
                                            const float* __restrict__ kv,
                                            float al[4], float ak[4]) {
#pragma unroll
  for (int i = 0; i < 4; ++i) {
    float a = 1.0f / (1.0f + expf(-ar[i]));
    al[i] = a;
    ak[i] = a * kv[i];
  }
}

// One-hot selector: inputs are exact {0,1} floats, so selecting the single
// non-zero term reproduces the reference dot products exactly.
// Left keys on chose_left; Right swaps the same/diff roles (index ^ 2).
__device__ __forceinline__ void step_coeffs(float cl, float o,
                                            const float al[4], const float ak[4],
                                            float& Al, float& Bl,
                                            float& Ar, float& Br) {
  const bool L = (cl != 0.0f);
  const bool O = (o  != 0.0f);
  const float a01 = O ? al[0] : al[1];   // same,  rew/unrew
  const float a23 = O ? al[2] : al[3];   // diff,  rew/unrew
  const float k01 = O ? ak[0] : ak[1];
  const float k23 = O ? ak[2] : ak[3];
  Al = 1.0f - (L ? a01 : a23);  Bl = L ? k01 : k23;
  Ar = 1.0f - (L ? a23 : a01);  Br = L ? k23 : k01;
}

__device__ __forceinline__ void unpack4(const v4f v0, const v4f v1, const v4f v2,
                                        float cl[4], float o[4]) {
  cl[0] = v0.x; o[0] = v0.z;
  cl[1] = v0.w; o[1] = v1.y;
  cl[2] = v1.z; o[2] = v2.x;
  cl[3] = v2.y; o[3] = v2.w;
}

// ---------------------------------------------------------------------------
// Fused kernel: one wave32 per session, one lane per 64-step chunk.
//  (1) async-DMA the lane's whole chunk strip into private LDS (ASYNCcnt)
//  (2) pass 1: per-chunk affine composition, consuming tiles as DMA lands
//  (3) wave shuffle scan composes chunk transforms -> exclusive start states
//  (4) pass 2: re-scan from LDS (no 2nd global read), non-temporal b128 stores
// ---------------------------------------------------------------------------
__global__ void __launch_bounds__(BLOCK)
qvalue_fused(const float* __restrict__ inp,
             const float* __restrict__ ar,
             const float* __restrict__ kv,
             float* __restrict__ out) {
  __shared__ __align__(16) char lds[LDS_BYTES];

  const int tid  = blockIdx.x * BLOCK + threadIdx.x;   // [0, NTASK)
  const int b    = tid >> 5;                           // session
  const int c    = tid & 31;                           // chunk == lane
  const unsigned lane = threadIdx.x & 31;

  const char* strip = (const char*)inp + ((size_t)b * T_TRIALS + (size_t)c * CH) * 12;
  char*       myl   = lds + (size_t)threadIdx.x * LANE_STRIDE_B;

  // (1) kick off the whole strip's DMA before doing any math
#pragma unroll
  for (int i = 0; i < N_ASYNC; ++i)
    async_copy16(strip + i * 16, myl + i * 16);

  float al[4], ak[4];
  load_params(ar, kv, al, ak);                         // overlaps with DMA

  // (2) pass 1: affine summary (Ap, Bc) with s_end = Ap*s0 + Bc
  const v4f* q = (const v4f*)myl;
  float ApL = 1.0f, BcL = 0.0f, ApR = 1.0f, BcR = 0.0f;
#pragma unroll
  for (int tile = 0; tile < 4; ++tile) {               // 16 steps per tile
    if      (tile == 0) wait_async<36>();              // oldest 12 copies done
    else if (tile == 1) wait_async<24>();
    else if (tile == 2) wait_async<12>();
    else                wait_async<0>();
#pragma unroll
    for (int g = tile * 4; g < tile * 4 + 4; ++g) {    // 4 steps / 48 B group
      const v4f v0 = q[3 * g + 0];
      const v4f v1 = q[3 * g + 1];
      const v4f v2 = q[3 * g + 2];
      float cl[4], o[4];
      unpack4(v0, v1, v2, cl, o);
#pragma unroll
      for (int t = 0; t < 4; ++t) {
        float Al, Bl, Ar, Br;
        step_coeffs(cl[t], o[t], al, ak, Al, Bl, Ar, Br);
        ApL *= Al; BcL = fmaf(Al, BcL, Bl);
        ApR *= Ar; BcR = fmaf(Ar, BcR, Br);
      }
    }
  }

  // (3) wave-level inclusive scan of affine transforms (oldest chunk first):
  //     compose(cur after prev): A = A_cur*A_prev, B = A_cur*B_prev + B_cur
  float aL = ApL, bL = BcL, aR = ApR, bR = BcR;
#pragma unroll
  for (int d = 1; d < 32; d <<= 1) {
    const float paL = __shfl_up(aL, d);
    const float pbL = __shfl_up(bL, d);
    const float paR = __shfl_up(aR, d);
    const float pbR = __shfl_up(bR, d);
    if (lane >= (unsigned)d) {
      bL = fmaf(aL, pbL, bL); aL *= paL;
      bR = fmaf(aR, pbR, bR); aR *= paR;
    }
  }
  // exclusive start state for this chunk (s0 = 0 -> start = B of lane-1)
  float sl = __shfl_up(bL, 1);
  float sr = __shfl_up(bR, 1);
  if (lane == 0) { sl = 0.0f; sr = 0.0f; }

  // (4) pass 2: re-scan from LDS, write every state (2 steps per b128 store)
  v4f* op = (v4f*)out + ((size_t)b * T_TRIALS + (size_t)c * CH) / 2;
#pragma unroll 4
  for (int g = 0; g < 16; ++g) {
    const v4f v0 = q[3 * g + 0];
    const v4f v1 = q[3 * g + 1];
    const v4f v2 = q[3 * g + 2];
    float cl[4], o[4];
    unpack4(v0, v1, v2, cl, o);

    v4f w01, w23;
    {
      float Al, Bl, Ar, Br;
      step_coeffs(cl[0], o[0], al, ak, Al, Bl, Ar, Br);
      sl = fmaf(Al, sl, Bl); sr = fmaf(Ar, sr, Br);
      w01.x = sl; w01.y = sr;
      step_coeffs(cl[1], o[1], al, ak, Al, Bl, Ar, Br);
      sl = fmaf(Al, sl, Bl); sr = fmaf(Ar, sr, Br);
      w01.z = sl; w01.w = sr;
      step_coeffs(cl[2], o[2], al, ak, Al, Bl, Ar, Br);
      sl = fmaf(Al, sl, Bl); sr = fmaf(Ar, sr, Br);
      w23.x = sl; w23.y = sr;
      step_coeffs(cl[3], o[3], al, ak, Al, Bl, Ar, Br);
      sl = fmaf(Al, sl, Bl); sr = fmaf(Ar, sr, Br);
      w23.z = sl; w23.w = sr;
    }
    // output is write-once/never-read: bypass cache residency (TH=NT)
    __builtin_nontemporal_store(w01, op + 2 * g + 0);
    __builtin_nontemporal_store(w23, op + 2 * g + 1);
  }
}

// ---------------------------------------------------------------------------
extern "C" void kernel_launch(void* const* d_in, const int* in_sizes, int n_in,
                              void* d_out, int out_size, void* d_ws, size_t ws_size,
                              hipStream_t stream) {
  const float* inp = (const float*)d_in[0];   // (4096, 2048, 3) f32
  const float* ar  = (const float*)d_in[1];   // (4,) f32
  const float* kv  = (const float*)d_in[2];   // (4,) f32
  float* out = (float*)d_out;                 // (4096, 2048, 2) f32
  (void)d_ws; (void)ws_size; (void)in_sizes; (void)n_in; (void)out_size;

  qvalue_fused<<<NTASK / BLOCK, BLOCK, 0, stream>>>(inp, ar, kv, out);
}